// HashEmbeddingEncoder_876173328999
// MI455X (gfx1250) — compile-verified
//
#include <hip/hip_runtime.h>
#include <hip/hip_bf16.h>
#include <stdint.h>

// Multiresolution hash-grid encoder (Instant-NGP style) for MI455X / gfx1250.
// One thread per (point, level); wave32 covers 2 points x 16 levels, giving
// fully coalesced 256B/wave float2 output stores. 8 corner embeddings are
// gathered with global_load_b64 from the L2-resident (64MB) tables.

typedef float v2f __attribute__((ext_vector_type(2)));

#define NUM_LEVELS 16
#define LOG2_T 19
#define TABLE_MASK ((1u << LOG2_T) - 1u)
#define PRIME_Y 2654435761u
#define PRIME_Z 805459861u

#define BLOCK 256
#define PTS_PER_BLOCK (BLOCK / NUM_LEVELS) /* 16 points per block */
#define XFLOATS (PTS_PER_BLOCK * 3)        /* 48 floats of coords per block */

__global__ __launch_bounds__(BLOCK) void HashEmbeddingEncoder_kernel(
    const float* __restrict__ x, const float* __restrict__ tables,
    float* __restrict__ out, int n) {
  __shared__ float sx[XFLOATS];
  const int t = threadIdx.x;

  // ---- Stage this block's 16 points (48 floats) into LDS via the gfx1250
  // ---- async load-to-LDS engine (ASYNCcnt-tracked), then barrier.
  {
    const int xi = blockIdx.x * XFLOATS + t;
    if (t < XFLOATS && xi < n * 3) {
      uint32_t lds_addr = (uint32_t)(uintptr_t)(void*)&sx[t];
      uint64_t gaddr = (uint64_t)(uintptr_t)(x + xi);
      asm volatile("global_load_async_to_lds_b32 %0, %1, off"
                   :
                   : "v"(lds_addr), "v"(gaddr)
                   : "memory");
    }
    // Issuing waves drain their async counter; waves that issued nothing see 0.
    asm volatile("s_wait_asynccnt 0" ::: "memory");
    __syncthreads();
  }

  const int gid = blockIdx.x * BLOCK + t;
  if (gid >= n * NUM_LEVELS) return;

  const int level = t & (NUM_LEVELS - 1);
  const int lp = t >> 4; // local point index within block

  // x = clip((x + 1) / 2, 0, 1 - 1e-6)   (SIZE == 1.0)
  const float cx_ = fminf(fmaxf((sx[lp * 3 + 0] + 1.0f) * 0.5f, 0.0f), 1.0f - 1e-6f);
  const float cy_ = fminf(fmaxf((sx[lp * 3 + 1] + 1.0f) * 0.5f, 0.0f), 1.0f - 1e-6f);
  const float cz_ = fminf(fmaxf((sx[lp * 3 + 2] + 1.0f) * 0.5f, 0.0f), 1.0f - 1e-6f);

  const float res = (float)(16 << level); // BASE_RES * 2^level
  const float sxc = cx_ * res, syc = cy_ * res, szc = cz_ * res;
  const float fxf = floorf(sxc), fyf = floorf(syc), fzf = floorf(szc);
  const float fx = sxc - fxf, fy = syc - fyf, fz = szc - fzf;
  const uint32_t ix = (uint32_t)fxf, iy = (uint32_t)fyf, iz = (uint32_t)fzf;

  // hash = (cx*1) ^ (cy*P1) ^ (cz*P2) ^ level ; fold level into x-term.
  const uint32_t hx0 = ix ^ (uint32_t)level;
  const uint32_t hx1 = (ix + 1u) ^ (uint32_t)level;
  const uint32_t hy0 = iy * PRIME_Y, hy1 = hy0 + PRIME_Y;
  const uint32_t hz0 = iz * PRIME_Z, hz1 = hz0 + PRIME_Z;

  const v2f* __restrict__ tab = (const v2f*)tables;
  const uint32_t lbase = (uint32_t)level << LOG2_T; // flat index into [L*T] table

  // Corner order (ox,oy,oz): 000,001,010,011,100,101,110,111
  uint32_t i0 = lbase + ((hx0 ^ hy0 ^ hz0) & TABLE_MASK);
  uint32_t i1 = lbase + ((hx0 ^ hy0 ^ hz1) & TABLE_MASK);
  uint32_t i2 = lbase + ((hx0 ^ hy1 ^ hz0) & TABLE_MASK);
  uint32_t i3 = lbase + ((hx0 ^ hy1 ^ hz1) & TABLE_MASK);
  uint32_t i4 = lbase + ((hx1 ^ hy0 ^ hz0) & TABLE_MASK);
  uint32_t i5 = lbase + ((hx1 ^ hy0 ^ hz1) & TABLE_MASK);
  uint32_t i6 = lbase + ((hx1 ^ hy1 ^ hz0) & TABLE_MASK);
  uint32_t i7 = lbase + ((hx1 ^ hy1 ^ hz1) & TABLE_MASK);

  // Issue all 8 b64 gathers before consuming (latency overlap in one clause).
  v2f e0 = tab[i0], e1 = tab[i1], e2 = tab[i2], e3 = tab[i3];
  v2f e4 = tab[i4], e5 = tab[i5], e6 = tab[i6], e7 = tab[i7];

  const float wx0 = 1.0f - fx, wx1 = fx;
  const float wy0 = 1.0f - fy, wy1 = fy;
  const float wz0 = 1.0f - fz, wz1 = fz;
  const float wyz00 = wy0 * wz0, wyz01 = wy0 * wz1;
  const float wyz10 = wy1 * wz0, wyz11 = wy1 * wz1;

  const float w0 = wx0 * wyz00, w1 = wx0 * wyz01, w2 = wx0 * wyz10, w3 = wx0 * wyz11;
  const float w4 = wx1 * wyz00, w5 = wx1 * wyz01, w6 = wx1 * wyz10, w7 = wx1 * wyz11;

  float ax = 0.0f, ay = 0.0f;
  ax = fmaf(w0, e0.x, ax); ay = fmaf(w0, e0.y, ay);
  ax = fmaf(w1, e1.x, ax); ay = fmaf(w1, e1.y, ay);
  ax = fmaf(w2, e2.x, ax); ay = fmaf(w2, e2.y, ay);
  ax = fmaf(w3, e3.x, ax); ay = fmaf(w3, e3.y, ay);
  ax = fmaf(w4, e4.x, ax); ay = fmaf(w4, e4.y, ay);
  ax = fmaf(w5, e5.x, ax); ay = fmaf(w5, e5.y, ay);
  ax = fmaf(w6, e6.x, ax); ay = fmaf(w6, e6.y, ay);
  ax = fmaf(w7, e7.x, ax); ay = fmaf(w7, e7.y, ay);

  // out[n, level*2 + d] -> float2 slot at gid; streamed (never re-read): NT store.
  v2f acc;
  acc.x = ax;
  acc.y = ay;
  __builtin_nontemporal_store(acc, (v2f*)out + gid);
}

extern "C" void kernel_launch(void* const* d_in, const int* in_sizes, int n_in,
                              void* d_out, int out_size, void* d_ws, size_t ws_size,
                              hipStream_t stream) {
  const float* x = (const float*)d_in[0];       // [N, 3] float32
  const float* tables = (const float*)d_in[1];  // [16, 2^19, 2] float32
  float* out = (float*)d_out;                   // [N, 32] float32

  const int n = in_sizes[0] / 3;
  const int total = n * NUM_LEVELS;
  const int blocks = (total + BLOCK - 1) / BLOCK;
  hipLaunchKernelGGL(HashEmbeddingEncoder_kernel, dim3(blocks), dim3(BLOCK), 0,
                     stream, x, tables, out, n);
}